// HausdorffDistanceLoss_74431783240208
// MI455X (gfx1250) — compile-verified
//
#include <hip/hip_runtime.h>

typedef __attribute__((ext_vector_type(2))) float v2f;
typedef __attribute__((ext_vector_type(4))) float f4;
typedef __attribute__((ext_vector_type(8))) float v8f;

#define RED_GRID  1024
#define RED_BLOCK 256

struct BlockPart { float s1; float s2; unsigned flags; unsigned pad; };

// per-element loss terms:
//   s1 += (p-t)^2 * (1-t)        [targets distance-transform branch]
//   s2 += (p-t)^2 * (1-pred)     [preds distance-transform branch]
// flags: bit0 = any(t==0), bit1 = any(pred==0)  (global "active" tests)
__device__ __forceinline__ void accum_elem(float x, float t,
                                           float& s1, float& s2,
                                           bool& anyT, bool& anyP) {
    float p  = 1.0f / (1.0f + __expf(-x));   // sigmoid
    float d  = p - t;
    float d2 = d * d;
    float invT = 1.0f - t;                    // binary targets -> 0/1
    bool  predZero = !(x > 0.0f);             // pred = (sigmoid(x) > 0.5) == (x > 0)
    s1 += d2 * invT;
    s2 += predZero ? d2 : 0.0f;
    anyT = anyT || (invT != 0.0f);
    anyP = anyP || predZero;
}

__global__ __launch_bounds__(RED_BLOCK)
void hausdorff_reduce(const float* __restrict__ logits,
                      const float* __restrict__ targets,
                      BlockPart* __restrict__ parts,
                      int nvec, int n) {
    const int tid    = blockIdx.x * blockDim.x + threadIdx.x;
    const int stride = gridDim.x * blockDim.x;

    float s1 = 0.0f, s2 = 0.0f;
    bool anyT = false, anyP = false;

    const f4* __restrict__ lg4 = (const f4*)logits;
    const f4* __restrict__ tg4 = (const f4*)targets;

    // Streaming pass: b128 non-temporal loads (single-touch data).
    for (int i = tid; i < nvec; i += stride) {
        f4 x4 = __builtin_nontemporal_load(&lg4[i]);
        f4 t4 = __builtin_nontemporal_load(&tg4[i]);
        accum_elem(x4.x, t4.x, s1, s2, anyT, anyP);
        accum_elem(x4.y, t4.y, s1, s2, anyT, anyP);
        accum_elem(x4.z, t4.z, s1, s2, anyT, anyP);
        accum_elem(x4.w, t4.w, s1, s2, anyT, anyP);
    }
    // scalar tail (n not multiple of 4)
    for (int i = 4 * nvec + tid; i < n; i += stride) {
        accum_elem(__builtin_nontemporal_load(&logits[i]),
                   __builtin_nontemporal_load(&targets[i]),
                   s1, s2, anyT, anyP);
    }

    // ---- wave32 cross-lane reduction via full-precision f32 WMMA ----
    const int lane = threadIdx.x & 31;
    float s1w, s2w;
#if __has_builtin(__builtin_amdgcn_wmma_f32_16x16x4_f32)
    {
        // A (16x4 f32): VGPR0 = {lanes0-15: K=0, lanes16-31: K=2},
        //               VGPR1 = {lanes0-15: K=1, lanes16-31: K=3}.
        // Put s1 in even-K slots (a.x), s2 in odd-K slots (a.y).
        v2f a; a.x = s1; a.y = s2;
        // B (4x16 f32): even-K rows feed columns 0-7 (s1), odd-K rows feed
        // columns 8-15 (s2). b.x covers K=0/2 (even), b.y covers K=1/3 (odd).
        float bx = ((lane & 15) < 8) ? 1.0f : 0.0f;
        v2f b; b.x = bx; b.y = 1.0f - bx;
        v8f c = {};
        v8f d = __builtin_amdgcn_wmma_f32_16x16x4_f32(
            /*neg_a=*/false, a, /*neg_b=*/false, b,
            /*c_mod=*/(short)0, c, /*reuse_a=*/false, /*reuse_b=*/false);
        // Sum the 8 D VGPRs: lane l<16 -> sum_{M=0..7} D[M,l];
        //                    lane l>=16 -> sum_{M=8..15} D[M,l-16].
        float acc = d[0] + d[1] + d[2] + d[3] + d[4] + d[5] + d[6] + d[7];
        // Column 0 carries s1 pair-sums, column 8 carries s2 pair-sums.
        s1w = __shfl(acc, 0, 32) + __shfl(acc, 16, 32);
        s2w = __shfl(acc, 8, 32) + __shfl(acc, 24, 32);
    }
#else
    {
        for (int off = 16; off; off >>= 1) {
            s1 += __shfl_xor(s1, off, 32);
            s2 += __shfl_xor(s2, off, 32);
        }
        s1w = s1; s2w = s2;
    }
#endif

    // ---- block reduction: 8 wave partials -> 1 per-block partial ----
    __shared__ float ls1[RED_BLOCK / 32];
    __shared__ float ls2[RED_BLOCK / 32];
    __shared__ unsigned lflags;
    if (threadIdx.x == 0) lflags = 0u;
    __syncthreads();
    if (lane == 0) {
        ls1[threadIdx.x >> 5] = s1w;
        ls2[threadIdx.x >> 5] = s2w;
    }
    if (anyT) atomicOr(&lflags, 1u);   // LDS ds_or, no-return
    if (anyP) atomicOr(&lflags, 2u);
    __syncthreads();
    if (threadIdx.x == 0) {
        float b1 = 0.0f, b2 = 0.0f;
        #pragma unroll
        for (int w = 0; w < RED_BLOCK / 32; ++w) { b1 += ls1[w]; b2 += ls2[w]; }
        BlockPart bp; bp.s1 = b1; bp.s2 = b2; bp.flags = lflags; bp.pad = 0u;
        parts[blockIdx.x] = bp;
    }
}

__global__ __launch_bounds__(256)
void hausdorff_finalize(const BlockPart* __restrict__ parts, int nparts,
                        float* __restrict__ out, double invN) {
    __shared__ double r1[256];
    __shared__ double r2[256];
    __shared__ unsigned rf[256];
    double a1 = 0.0, a2 = 0.0; unsigned f = 0u;
    for (int i = threadIdx.x; i < nparts; i += 256) {
        BlockPart bp = parts[i];
        a1 += (double)bp.s1;
        a2 += (double)bp.s2;
        f  |= bp.flags;
    }
    r1[threadIdx.x] = a1; r2[threadIdx.x] = a2; rf[threadIdx.x] = f;
    __syncthreads();
    for (int s = 128; s > 0; s >>= 1) {
        if (threadIdx.x < (unsigned)s) {
            r1[threadIdx.x] += r1[threadIdx.x + s];
            r2[threadIdx.x] += r2[threadIdx.x + s];
            rf[threadIdx.x] |= rf[threadIdx.x + s];
        }
        __syncthreads();
    }
    if (threadIdx.x == 0) {
        double sum = ((rf[0] & 1u) ? r1[0] : 0.0) + ((rf[0] & 2u) ? r2[0] : 0.0);
        out[0] = (float)(sum * invN);
    }
}

extern "C" void kernel_launch(void* const* d_in, const int* in_sizes, int n_in,
                              void* d_out, int out_size, void* d_ws, size_t ws_size,
                              hipStream_t stream) {
    (void)n_in; (void)out_size; (void)ws_size;
    const float* logits  = (const float*)d_in[0];
    const float* targets = (const float*)d_in[1];
    float* out = (float*)d_out;
    BlockPart* parts = (BlockPart*)d_ws;   // RED_GRID * 16 bytes = 16 KB scratch

    const int n    = in_sizes[0];
    const int nvec = n >> 2;

    hausdorff_reduce<<<RED_GRID, RED_BLOCK, 0, stream>>>(logits, targets, parts, nvec, n);
    hausdorff_finalize<<<1, 256, 0, stream>>>(parts, RED_GRID, out, 1.0 / (double)n);
}